// DPConv_16011638080100
// MI455X (gfx1250) — compile-verified
//
#include <hip/hip_runtime.h>
#include <hip/hip_bf16.h>

typedef float v2f __attribute__((ext_vector_type(2)));
typedef float v8f __attribute__((ext_vector_type(8)));
typedef unsigned int u32x4 __attribute__((ext_vector_type(4)));
typedef int i32x4 __attribute__((ext_vector_type(4)));
typedef int i32x8 __attribute__((ext_vector_type(8)));

// ---------------------------------------------------------------------------
// TDM helper: DMA a contiguous run of `nelem` f32 from global `src` into LDS
// at byte offset `lds_off`. Descriptor per CDNA5 ISA:
//   group0: count=1 | lds_addr | global_addr[56:0] | type=2
//   group1: data_size=2 (4B), tensor_dim0 = tile_dim0 = nelem, dim1 = 1,
//           tensor_dim0_stride = nelem
//   group2/3: 1-deep / unused (tile_dim3/4 = 0)
// This toolchain exposes the 6-arg builtin: (v4u, v8i, v4i, v4i, v8i, i32).
// Issued by the calling wave only; TDM ignores EXEC. Caller must barrier.
// ---------------------------------------------------------------------------
#if __has_builtin(__builtin_amdgcn_tensor_load_to_lds)
#define HAVE_TDM 1
__device__ __forceinline__ void tdm_load_f32(unsigned lds_off,
                                             const float* src,
                                             unsigned nelem) {
    const unsigned long long ga = (unsigned long long)(uintptr_t)src;
    u32x4 g0;
    g0[0] = 1u;                                        // count=1, user mode
    g0[1] = lds_off;                                   // LDS byte address
    g0[2] = (unsigned)(ga & 0xffffffffu);              // global_addr[31:0]
    g0[3] = (unsigned)((ga >> 32) & 0x01ffffffu)       // global_addr[56:32]
          | (2u << 30);                                // type = 2 ("image")
    i32x8 g1;
    g1[0] = (int)(2u << 16);                           // data_size = 4 bytes
    g1[1] = (int)((nelem & 0xffffu) << 16);            // tensor_dim0[15:0]
    g1[2] = (int)((nelem >> 16) | (1u << 16));         // dim0[31:16], dim1=1
    g1[3] = (int)((nelem & 0xffffu) << 16);            // tile_dim0 = nelem
    g1[4] = 1;                                         // tile_dim1=1, tile_dim2=0
    g1[5] = (int)nelem;                                // tensor_dim0_stride lo
    g1[6] = 0;                                         // stride hi, dim1_stride lo
    g1[7] = 0;
    i32x4 g2; g2[0] = 1; g2[1] = 0; g2[2] = 0; g2[3] = 0;   // tensor_dim2=1
    i32x4 g3; g3[0] = 0; g3[1] = 0; g3[2] = 0; g3[3] = 0;
    i32x8 g4;
    g4[0] = 0; g4[1] = 0; g4[2] = 0; g4[3] = 0;
    g4[4] = 0; g4[5] = 0; g4[6] = 0; g4[7] = 0;
    __builtin_amdgcn_tensor_load_to_lds(g0, g1, g2, g3, g4, 0);
}
#else
#define HAVE_TDM 0
#endif

// ---------------------------------------------------------------------------
// Generic 1x1 conv as GEMM:  out[img, co, hw] = sum_ci W[co,ci] * in[img,ci,hw] + b[co]
// One wave handles 16 consecutive pixels of one image and ALL Cout tiles
// (reuses the B fragment across Cout tiles). Weights staged in LDS via TDM.
// WMMA f32 16x16x4 layouts (ISA 7.12.2):
//   A 16x4: lane half0 -> K = k0,k0+1 ; half1 -> K = k0+2,k0+3
//   B 4x16: same half split, N = lane%16
//   C/D 16x16: VGPR r -> M = r + 8*half, N = lane%16
// ---------------------------------------------------------------------------
template<int CIN, int COUT>
__global__ __launch_bounds__(256) void conv1x1_wmma(
    const float* __restrict__ in,    // (B, CIN, HW)
    const float* __restrict__ w,     // (COUT, CIN)
    const float* __restrict__ bias,  // (COUT)
    float* __restrict__ out,         // (B, COUT, HW)
    int HW)
{
    __shared__ __align__(16) float sw[COUT * CIN];
#if HAVE_TDM
    if (threadIdx.y == 0) {
        tdm_load_f32((unsigned)(uintptr_t)(void*)sw, w, COUT * CIN);
        __builtin_amdgcn_s_wait_tensorcnt(0);
    }
#else
    for (int i = threadIdx.y * 32 + threadIdx.x; i < COUT * CIN; i += 256)
        sw[i] = w[i];
#endif
    __syncthreads();

    const int group = blockIdx.x * 8 + threadIdx.y;   // 16-pixel group id
    const int gper  = HW / 16;                        // groups per image
    const int img   = group / gper;
    const int hw0   = (group - img * gper) * 16;
    const int lane  = threadIdx.x;
    const int half  = lane >> 4;        // 0: lanes 0-15, 1: lanes 16-31
    const int mn    = lane & 15;        // M (for A) / N (for B)

    const float* inbase = in + (size_t)img * CIN * HW + hw0;

    constexpr int NT = COUT / 16;
    v8f acc[NT];
#pragma unroll
    for (int t = 0; t < NT; ++t)
#pragma unroll
        for (int r = 0; r < 8; ++r) acc[t][r] = 0.0f;

    for (int kk = 0; kk < CIN / 4; ++kk) {
        const int k0 = kk * 4 + 2 * half;
        v2f bf;
        bf[0] = inbase[(size_t)(k0 + 0) * HW + mn];
        bf[1] = inbase[(size_t)(k0 + 1) * HW + mn];
        if (kk + 1 < CIN / 4)
            __builtin_prefetch(inbase + (size_t)(kk * 4 + 4 + 2 * half) * HW + mn, 0, 0);
#pragma unroll
        for (int t = 0; t < NT; ++t) {
            const int co = t * 16 + mn;
            v2f af;
            af[0] = sw[co * CIN + k0 + 0];
            af[1] = sw[co * CIN + k0 + 1];
            acc[t] = __builtin_amdgcn_wmma_f32_16x16x4_f32(
                false, af, false, bf, (short)0, acc[t], false, false);
        }
    }

#pragma unroll
    for (int t = 0; t < NT; ++t) {
#pragma unroll
        for (int r = 0; r < 8; ++r) {
            const int co = t * 16 + r + 8 * half;
            out[((size_t)img * COUT + co) * HW + hw0 + mn] = acc[t][r] + bias[co];
        }
    }
}

// ---------------------------------------------------------------------------
// Adaptive pool 34x34 -> 32x32 over 4x4 edge-padded windows of x1 (= y ch 0..63).
// pooled: (256, 64, 32, 32). Each output pixel touches at most 2x2 inputs.
// ---------------------------------------------------------------------------
__global__ __launch_bounds__(256) void pool_kernel(
    const float* __restrict__ y,      // (16,128,128,128)
    float* __restrict__ pooled)       // (256,64,32,32)
{
    const int idx = blockIdx.x * 256 + threadIdx.x;   // 256*64*1024 total
    const int pix = idx & 1023;
    const int c   = (idx >> 10) & 63;
    const int win = idx >> 16;
    const int n   = win >> 4;
    const int wi  = (win >> 2) & 3;
    const int wj  = win & 3;
    const int Hh  = pix >> 5;
    const int Ww  = pix & 31;

    // adaptive_matrix(34,32): row i covers [i*34//32, ceil((i+1)*34/32))
    const int hs = (Hh * 34) >> 5, he = ((Hh + 1) * 34 + 31) >> 5;
    const int vs = (Ww * 34) >> 5, ve = ((Ww + 1) * 34 + 31) >> 5;

    const float* plane = y + ((size_t)n * 128 + c) * 16384;
    float sum = 0.0f;
    for (int h = hs; h < he; ++h) {
        int hp = wi * 34 + h - 4;                 // undo edge pad of 4
        hp = hp < 0 ? 0 : (hp > 127 ? 127 : hp);
        for (int v = vs; v < ve; ++v) {
            int wp = wj * 34 + v - 4;
            wp = wp < 0 ? 0 : (wp > 127 ? 127 : wp);
            sum += plane[hp * 128 + wp];
        }
    }
    pooled[idx] = sum / (float)((he - hs) * (ve - vs));
}

// ---------------------------------------------------------------------------
// pr = conv2(1x1) + b2 + pos(3x3 SAME) + bp  on pooled (256,64,32,32).
// Fused as a 10-tap WMMA GEMM: tap 0 = conv2 (center), taps 1..9 = pos taps.
// Block = 4 output rows of one window; the 6x34x64 input tile (zero halo) is
// staged in LDS with channel-contiguous layout so every B fragment is one
// unconditional ds_load_b64 -- no EXEC predication in the inner loop.
// ---------------------------------------------------------------------------
__global__ __launch_bounds__(256) void conv2pos_wmma(
    const float* __restrict__ pooled, // (256,64,32,32)
    const float* __restrict__ w2,     // (64,64)
    const float* __restrict__ b2,     // (64)
    const float* __restrict__ wp,     // (64,64,3,3)
    const float* __restrict__ bp,     // (64)
    float* __restrict__ pr)           // (256,64,32,32)
{
    __shared__ __align__(16) float tile[6 * 34 * 64];   // 52224 B

    const int win = blockIdx.x >> 3;        // 256 windows
    const int hh0 = (blockIdx.x & 7) * 4;   // first output row of this block
    const float* pb = pooled + (size_t)win * 65536;

    // Stage rows hh0-1 .. hh0+4, cols -1..32, all 64 channels; zeros outside.
    const int tid = threadIdx.y * 32 + threadIdx.x;
    for (int e = tid; e < 6 * 34 * 64; e += 256) {
        const int ci = e & 63;
        const int rc = e >> 6;              // 0..203
        const int col = rc % 34;            // gw = col-1
        const int r   = rc / 34;            // gh = hh0-1+r
        const int gh = hh0 - 1 + r;
        const int gw = col - 1;
        float v = 0.0f;
        if ((unsigned)gh < 32u && (unsigned)gw < 32u)
            v = pb[(size_t)ci * 1024 + gh * 32 + gw];
        tile[e] = v;
    }
    __syncthreads();

    const int hh = hh0 + (threadIdx.y >> 1);
    const int w0 = (threadIdx.y & 1) * 16;
    const int lane = threadIdx.x;
    const int half = lane >> 4;
    const int mn   = lane & 15;

    v8f acc[4];
#pragma unroll
    for (int t = 0; t < 4; ++t)
#pragma unroll
        for (int r = 0; r < 8; ++r) acc[t][r] = 0.0f;

    for (int tap = 0; tap < 10; ++tap) {
        const int ky = (tap == 0) ? 1 : ((tap - 1) / 3);
        const int kx = (tap == 0) ? 1 : ((tap - 1) % 3);
        // branch-free weight addressing: w2 stride 1, pos stride 9 (+tap offset)
        const float* wbase = (tap == 0) ? w2 : (wp + ky * 3 + kx);
        const int wstep = (tap == 0) ? 1 : 9;
        const int r  = (hh - hh0) + ky;          // 0..5  (row in tile)
        const int cc = w0 + mn + kx;             // 0..33 (col in tile)
        const float* bsrc = &tile[(r * 34 + cc) * 64];

        for (int kk = 0; kk < 16; ++kk) {
            const int k0 = kk * 4 + 2 * half;
            const v2f bf = *(const v2f*)(bsrc + k0);   // ds_load_b64
#pragma unroll
            for (int t = 0; t < 4; ++t) {
                const int co = t * 16 + mn;
                v2f af;
                af[0] = wbase[(co * 64 + k0 + 0) * wstep];
                af[1] = wbase[(co * 64 + k0 + 1) * wstep];
                acc[t] = __builtin_amdgcn_wmma_f32_16x16x4_f32(
                    false, af, false, bf, (short)0, acc[t], false, false);
            }
        }
    }

#pragma unroll
    for (int t = 0; t < 4; ++t) {
#pragma unroll
        for (int r = 0; r < 8; ++r) {
            const int co = t * 16 + r + 8 * half;
            pr[(size_t)win * 65536 + (size_t)co * 1024 + hh * 32 + w0 + mn] =
                acc[t][r] + b2[co] + bp[co];
        }
    }
}

// ---------------------------------------------------------------------------
// SE block: per-window channel means -> 8-dim bottleneck -> sigmoid scale.
// sbuf[win*64 + c] = sigmoid(se_w2 @ relu(se_w1 @ mean))
// ---------------------------------------------------------------------------
__global__ __launch_bounds__(256) void se_kernel(
    const float* __restrict__ pr,     // (256,64,32,32)
    const float* __restrict__ w1,     // (8,64)
    const float* __restrict__ w2,     // (64,8)
    float* __restrict__ sbuf)         // (256,64)
{
    __shared__ float red[256];
    __shared__ float mean[64];
    __shared__ float t1[8];

    const int win = blockIdx.x;
    const int t = threadIdx.x;
    const int c = t >> 2, part = t & 3;

    const float* p = pr + (size_t)win * 65536 + (size_t)c * 1024 + part * 256;
    float sacc = 0.0f;
    for (int i = 0; i < 256; ++i) sacc += p[i];
    red[t] = sacc;
    __syncthreads();
    if (part == 0)
        mean[c] = (red[t] + red[t + 1] + red[t + 2] + red[t + 3]) * (1.0f / 1024.0f);
    __syncthreads();
    if (t < 8) {
        float a = 0.0f;
        for (int cc = 0; cc < 64; ++cc) a += w1[t * 64 + cc] * mean[cc];
        t1[t] = a > 0.0f ? a : 0.0f;
    }
    __syncthreads();
    if (t < 64) {
        float a = 0.0f;
        for (int cr = 0; cr < 8; ++cr) a += w2[t * 8 + cr] * t1[cr];
        sbuf[(size_t)win * 64 + t] = 1.0f / (1.0f + expf(-a));
    }
}

// ---------------------------------------------------------------------------
// Gather + concat: catin (16,128,128,128).
//   c <  64 : flat-reinterpret scramble of att = pr*(1+s):
//             f = c*16384 + h'*512 + w'*16 + (4i+j) ; lw=f>>16, cc, pix
//   c >= 64 : copy x2 = y channels 64..127
// (H%nw==0, so the reference's overlap-add is an exact tiling; cnt==1.)
// ---------------------------------------------------------------------------
__global__ __launch_bounds__(256) void gather_kernel(
    const float* __restrict__ pr,     // (256,64,32,32)
    const float* __restrict__ sbuf,   // (256,64)
    const float* __restrict__ y,      // (16,128,128,128)
    float* __restrict__ catin)        // (16,128,128,128)
{
    const int idx = blockIdx.x * 256 + threadIdx.x;   // 16*128*16384 total
    const int hw = idx & 16383;
    const int c  = (idx >> 14) & 127;
    const int n  = idx >> 21;

    float v;
    if (c < 64) {
        const int h = hw >> 7, w = hw & 127;
        const int i = h >> 5, j = w >> 5;
        const int hp = h & 31, wp = w & 31;
        const int f = c * 16384 + hp * 512 + wp * 16 + (i * 4 + j);
        const int lw  = f >> 16;
        const int rem = f & 65535;
        const int cc  = rem >> 10;
        const int pix = rem & 1023;
        const size_t wwin = (size_t)(n * 16 + lw);
        const float prv = pr[wwin * 65536 + (size_t)cc * 1024 + pix];
        const float sc  = sbuf[wwin * 64 + cc];
        v = prv * (1.0f + sc);
    } else {
        v = y[((size_t)n * 128 + c) * 16384 + hw];
    }
    catin[idx] = v;
}

// ---------------------------------------------------------------------------
extern "C" void kernel_launch(void* const* d_in, const int* in_sizes, int n_in,
                              void* d_out, int out_size, void* d_ws, size_t ws_size,
                              hipStream_t stream) {
    const float* x       = (const float*)d_in[0];
    const float* conv1_w = (const float*)d_in[1];
    const float* conv1_b = (const float*)d_in[2];
    const float* conv2_w = (const float*)d_in[3];
    const float* conv2_b = (const float*)d_in[4];
    const float* conv3_w = (const float*)d_in[5];
    const float* conv3_b = (const float*)d_in[6];
    const float* pos_w   = (const float*)d_in[7];
    const float* pos_b   = (const float*)d_in[8];
    const float* se_w1   = (const float*)d_in[9];
    const float* se_w2   = (const float*)d_in[10];

    float* ws     = (float*)d_ws;
    float* y      = ws;                    // (16,128,128,128)  33554432 f
    float* pooled = y + 33554432;          // (256,64,32,32)    16777216 f
    float* pr     = pooled + 16777216;     // (256,64,32,32)    16777216 f
    float* sbuf   = pr + 16777216;         // (256,64)             16384 f
    float* catin  = sbuf + 16384;          // (16,128,128,128)  33554432 f

    dim3 wblock(32, 8);

    // conv1: (64 -> 128), 16384 pixel-groups, 8 waves/block
    conv1x1_wmma<64, 128><<<2048, wblock, 0, stream>>>(x, conv1_w, conv1_b, y, 16384);

    // adaptive pool of x1 windows
    pool_kernel<<<65536, 256, 0, stream>>>(y, pooled);

    // fused conv2 + 3x3 pos conv (LDS-staged tiles)
    conv2pos_wmma<<<2048, wblock, 0, stream>>>(pooled, conv2_w, conv2_b, pos_w, pos_b, pr);

    // SE scales
    se_kernel<<<256, 256, 0, stream>>>(pr, se_w1, se_w2, sbuf);

    // scramble-gather (+SE apply) and concat with x2
    gather_kernel<<<131072, 256, 0, stream>>>(pr, sbuf, y, catin);

    // conv3: (128 -> 64) -> output
    conv1x1_wmma<128, 64><<<2048, wblock, 0, stream>>>(catin, conv3_w, conv3_b,
                                                       (float*)d_out, 16384);
}